// Model_5351529250912
// MI455X (gfx1250) — compile-verified
//
#include <hip/hip_runtime.h>
#include <math.h>
#include <stdint.h>

#define T_STEPS 600
#define N_REC   5000
#define N_IN    100
#define N_OUT   100
#define ALPHA   0.1f

typedef __attribute__((ext_vector_type(2))) float v2f;
typedef __attribute__((ext_vector_type(8))) float v8f;

// ---------------------------------------------------------------------------
// x0 = arctanh(r); hidden[0] = r
// ---------------------------------------------------------------------------
__global__ void init_state(const float* __restrict__ r,
                           float* __restrict__ x,
                           float* __restrict__ hidden0) {
  int i = blockIdx.x * blockDim.x + threadIdx.x;
  if (i < N_REC) {
    float rv = r[i];
    x[i] = atanhf(rv);
    hidden0[i] = rv;
  }
}

// ---------------------------------------------------------------------------
// C[M,N] = A[M,K] @ Wt[N,K]^T  (+ E[M,N] if E != nullptr)
// One 16x16 C-tile per wave, K-loop in steps of 4 using V_WMMA_F32_16X16X4_F32.
// f32 end-to-end (matches reference precision).
//
// Edge handling by CLAMPING row/col indices (not predication): an out-of-range
// A-row / B-col only contributes to out-of-range C entries, which are never
// stored. K is a multiple of 4 for both call sites, so the inner loop is
// branch-free: one b64 load per fragment + one wmma.
// ---------------------------------------------------------------------------
__global__ __launch_bounds__(128) void gemm_wmma_f32(
    const float* __restrict__ A, const float* __restrict__ Wt,
    const float* __restrict__ E, float* __restrict__ C,
    int M, int N, int K) {
  int wave   = (blockIdx.x * blockDim.x + threadIdx.x) >> 5;
  int lane   = threadIdx.x & 31;
  int mTiles = (M + 15) >> 4;
  int nTiles = (N + 15) >> 4;
  if (wave >= mTiles * nTiles) return;   // uniform per wave: EXEC stays all-ones

  int mT = wave / nTiles;
  int nT = wave - mT * nTiles;
  int mBase = mT << 4;
  int nBase = nT << 4;
  int half  = lane >> 4;     // 0 or 1
  int lr    = lane & 15;

  int aRow = min(mBase + lr, M - 1);     // clamped: safe, see header comment
  int bCol = min(nBase + lr, N - 1);

  const v2f* __restrict__ Ap =
      reinterpret_cast<const v2f*>(A + (size_t)aRow * K + (half << 1));
  const v2f* __restrict__ Bp =
      reinterpret_cast<const v2f*>(Wt + (size_t)bCol * K + (half << 1));

  v8f c;
  if (E != nullptr) {
#pragma unroll
    for (int p = 0; p < 8; ++p) {
      int row = min(mBase + p + (half << 3), M - 1);
      c[p] = E[(size_t)row * N + bCol];
    }
  } else {
#pragma unroll
    for (int p = 0; p < 8; ++p) c[p] = 0.0f;
  }

  int kSteps = K >> 2;                   // K % 4 == 0 at both call sites
#pragma unroll 4
  for (int s = 0; s < kSteps; ++s) {
    v2f a = Ap[s << 1];
    v2f b = Bp[s << 1];
    // (neg_a, A, neg_b, B, c_mod, C, reuse_a, reuse_b)
    c = __builtin_amdgcn_wmma_f32_16x16x4_f32(false, a, false, b,
                                              (short)0, c, false, false);
  }

#pragma unroll
  for (int p = 0; p < 8; ++p) {
    int row = mBase + p + (half << 3);
    int col = nBase + lr;
    if (row < M && col < N) C[(size_t)row * N + col] = c[p];
  }
}

// ---------------------------------------------------------------------------
// One recurrence step: for every row i,
//   acc = W_rec[i,:] . r_prev  (+ W_in[i,:] . u_t if fused-drive path)
//   x[i] = (1-ALPHA)*x[i] + ALPHA*(acc + bias)   bias = drive_t[i] or eps_t[i]
//   r_new[i] = tanh(x[i])
// 256 threads = 8 waves, one row per wave, float4 coalesced row reads,
// r_prev staged into LDS via gfx1250 GLOBAL_LOAD_ASYNC_TO_LDS_B128 (ASYNCcnt,
// no VGPR round-trip), wave32 shuffle reduction.
// W_rec (100 MB) is L2-resident on MI455X (192 MB L2) -> runs at L2 bandwidth.
// ---------------------------------------------------------------------------
__global__ __launch_bounds__(256) void rnn_step(
    const float* __restrict__ Wrec,
    const float* __restrict__ r_prev,
    float* __restrict__ r_new,
    float* __restrict__ x,
    const float* __restrict__ drive_t,   // precomputed drive row (or null)
    const float* __restrict__ Win,
    const float* __restrict__ u_t,
    const float* __restrict__ eps_t,
    int fused) {
  __shared__ float4 s_r[N_REC / 4];      // 20 KB of 320 KB LDS
  __shared__ float4 s_u[N_IN / 4];

  int tid = threadIdx.x;

  // Async-copy r_prev into LDS: each lane moves 16 B directly global->LDS.
  // Generic address of a __shared__ object carries the LDS byte offset in its
  // low 32 bits (LDS aperture rule).
  {
    uint32_t lds0 = (uint32_t)(uintptr_t)(&s_r[0]);
    const float4* r4 = reinterpret_cast<const float4*>(r_prev);
    for (int i = tid; i < N_REC / 4; i += 256) {
      uint32_t la = lds0 + (uint32_t)i * 16u;
      uint64_t ga = (uint64_t)(uintptr_t)(r4 + i);
      asm volatile("global_load_async_to_lds_b128 %0, %1, off"
                   :: "v"(la), "v"(ga) : "memory");
    }
  }
  if (fused && tid < N_IN / 4)
    s_u[tid] = reinterpret_cast<const float4*>(u_t)[tid];
  asm volatile("s_wait_asynccnt 0x0" ::: "memory");
  __syncthreads();

  int wv   = tid >> 5;
  int lane = tid & 31;
  int row  = blockIdx.x * 8 + wv;        // grid = N_REC/8 exactly

  const float4* w4 = reinterpret_cast<const float4*>(Wrec + (size_t)row * N_REC);
  float acc = 0.0f;
  for (int j = lane; j < N_REC / 4; j += 32) {
    float4 w  = w4[j];
    float4 rr = s_r[j];
    acc = fmaf(w.x, rr.x, acc);
    acc = fmaf(w.y, rr.y, acc);
    acc = fmaf(w.z, rr.z, acc);
    acc = fmaf(w.w, rr.w, acc);
  }
  if (fused) {
    const float4* wi4 = reinterpret_cast<const float4*>(Win + (size_t)row * N_IN);
    for (int j = lane; j < N_IN / 4; j += 32) {
      float4 w  = wi4[j];
      float4 uu = s_u[j];
      acc = fmaf(w.x, uu.x, acc);
      acc = fmaf(w.y, uu.y, acc);
      acc = fmaf(w.z, uu.z, acc);
      acc = fmaf(w.w, uu.w, acc);
    }
  }
#pragma unroll
  for (int off = 16; off > 0; off >>= 1)
    acc += __shfl_xor(acc, off, 32);

  if (lane == 0) {
    float bias = fused ? eps_t[row] : drive_t[row];
    float xn = (1.0f - ALPHA) * x[row] + ALPHA * (acc + bias);
    x[row] = xn;
    r_new[row] = tanhf(xn);
  }
}

// ---------------------------------------------------------------------------
extern "C" void kernel_launch(void* const* d_in, const int* in_sizes, int n_in,
                              void* d_out, int out_size, void* d_ws, size_t ws_size,
                              hipStream_t stream) {
  const float* u    = (const float*)d_in[0];   // [600,100]
  const float* r    = (const float*)d_in[1];   // [5000]
  const float* Win  = (const float*)d_in[2];   // [5000,100]
  const float* Wrec = (const float*)d_in[3];   // [5000,5000]
  const float* Wout = (const float*)d_in[4];   // [100,5000]
  const float* eps  = (const float*)d_in[5];   // [600,5000]

  float* hidden = (float*)d_out;                          // [601,5000]
  float* o      = hidden + (size_t)(T_STEPS + 1) * N_REC; // [600,100]

  float* x     = (float*)d_ws;                            // [5000]
  float* drive = x + N_REC;                               // [600,5000] if it fits
  size_t ws_needed = (size_t)(N_REC + (size_t)T_STEPS * N_REC) * sizeof(float);
  bool useDrive = (ws_size >= ws_needed);

  init_state<<<(N_REC + 255) / 256, 256, 0, stream>>>(r, x, hidden);

  if (useDrive) {
    // drive = u @ W_in^T + eps  (WMMA f32)
    int tiles = ((T_STEPS + 15) / 16) * ((N_REC + 15) / 16);
    gemm_wmma_f32<<<(tiles + 3) / 4, 128, 0, stream>>>(
        u, Win, eps, drive, T_STEPS, N_REC, N_IN);
  }

  for (int t = 0; t < T_STEPS; ++t) {
    rnn_step<<<N_REC / 8, 256, 0, stream>>>(
        Wrec,
        hidden + (size_t)t * N_REC,
        hidden + (size_t)(t + 1) * N_REC,
        x,
        useDrive ? (drive + (size_t)t * N_REC) : nullptr,
        Win,
        u + (size_t)t * N_IN,
        eps + (size_t)t * N_REC,
        useDrive ? 0 : 1);
  }

  // o = rs @ W_out^T  (WMMA f32), rs = hidden rows 1..600
  int tiles2 = ((T_STEPS + 15) / 16) * ((N_OUT + 15) / 16);
  gemm_wmma_f32<<<(tiles2 + 3) / 4, 128, 0, stream>>>(
      hidden + N_REC, Wout, nullptr, o, T_STEPS, N_OUT, N_REC);
}